// BiLSTMEncoder_25786983645879
// MI455X (gfx1250) — compile-verified
//
#include <hip/hip_runtime.h>
#include <hip/hip_bf16.h>

typedef float v2f __attribute__((ext_vector_type(2)));
typedef float v8f __attribute__((ext_vector_type(8)));

#define WMMA_F32(a, b, c) \
    __builtin_amdgcn_wmma_f32_16x16x4_f32(false, (a), false, (b), (short)0, (c), false, false)

#define EDIM 768
#define HDIM 384
#define G4H  1536
#define BSZ  32
#define SSZ  512
#define ROWS (BSZ * SSZ)   // 16384

__device__ __forceinline__ float sigmoid_f(float x) {
    return 1.0f / (1.0f + __expf(-x));
}
__device__ __forceinline__ float tanh_f(float x) {
    // tanh(x) = 1 - 2/(exp(2x)+1)
    return 1.0f - 2.0f / (__expf(2.0f * x) + 1.0f);
}

// ---------------------------------------------------------------------------
// Kernel A: xg[dir][s][b][n] = A[dir_row] @ Wih^T + (bih + bhh)
//   A rows indexed r = b*512 + s ; for layer0 backward dir, s is time-reversed.
//   Block tile 64(M) x 64(N), 128 threads = 4 waves, wave tile 32x32.
// ---------------------------------------------------------------------------
__global__ __launch_bounds__(128)
void xg_gemm_kernel(const float* __restrict__ Af, const float* __restrict__ Ab,
                    const float* __restrict__ Wf, const float* __restrict__ Wb,
                    const float* __restrict__ bihf, const float* __restrict__ bhhf,
                    const float* __restrict__ bihb, const float* __restrict__ bhhb,
                    float* __restrict__ xg, int Kin, int reverse)
{
    const int dir = blockIdx.z;
    const float* __restrict__ A   = dir ? Ab   : Af;
    const float* __restrict__ W   = dir ? Wb   : Wf;
    const float* __restrict__ bih = dir ? bihb : bihf;
    const float* __restrict__ bhh = dir ? bhhb : bhhf;

    const int n0 = blockIdx.x * 64;
    const int m0 = blockIdx.y * 64;

    __shared__ float As[64][17];   // [m][k], pad -> conflict-free
    __shared__ float Bs[16][65];   // [k][n], pad -> conflict-free

    const int tid   = threadIdx.x;
    const int lane  = tid & 31;
    const int wid   = tid >> 5;          // 0..3
    const int wm    = (wid & 1) * 32;    // wave M offset in block
    const int wn    = (wid >> 1) * 32;   // wave N offset in block
    const int lhalf = lane >> 4;         // 0/1
    const int l16   = lane & 15;

    v8f acc[2][2] = {};

    for (int k0 = 0; k0 < Kin; k0 += 16) {
        // load A tile 64x16 (per-row time reversal for backward layer-0)
        #pragma unroll
        for (int i = 0; i < 8; i++) {
            int idx = tid + i * 128;
            int mr  = idx >> 4;
            int kk  = idx & 15;
            int m   = m0 + mr;
            int b   = m >> 9;
            int s   = m & 511;
            int srd = (reverse && dir) ? (511 - s) : s;
            As[mr][kk] = A[(size_t)(b * SSZ + srd) * Kin + (k0 + kk)];
        }
        // load B tile: Bs[k][n] = W[n0+n][k0+k]
        #pragma unroll
        for (int i = 0; i < 8; i++) {
            int idx = tid + i * 128;
            int nr  = idx >> 4;
            int kk  = idx & 15;
            Bs[kk][nr] = W[(size_t)(n0 + nr) * Kin + (k0 + kk)];
        }
        __syncthreads();

        #pragma unroll
        for (int kc = 0; kc < 16; kc += 4) {
            v2f afrag[2], bfrag[2];
            #pragma unroll
            for (int mt = 0; mt < 2; mt++) {
                int m = wm + mt * 16 + l16;
                afrag[mt].x = As[m][kc + 2 * lhalf];
                afrag[mt].y = As[m][kc + 2 * lhalf + 1];
            }
            #pragma unroll
            for (int nt = 0; nt < 2; nt++) {
                int n = wn + nt * 16 + l16;
                bfrag[nt].x = Bs[kc + 2 * lhalf][n];
                bfrag[nt].y = Bs[kc + 2 * lhalf + 1][n];
            }
            #pragma unroll
            for (int mt = 0; mt < 2; mt++)
                #pragma unroll
                for (int nt = 0; nt < 2; nt++)
                    acc[mt][nt] = WMMA_F32(afrag[mt], bfrag[nt], acc[mt][nt]);
        }
        __syncthreads();
    }

    // epilogue: add biases, store time-major xg[dir][s][b][1536].
    // Streaming data (write once, read once): non-temporal stores keep L2
    // lines free for the weight matrices the recurrence re-reads 512x.
    #pragma unroll
    for (int mt = 0; mt < 2; mt++)
        #pragma unroll
        for (int nt = 0; nt < 2; nt++) {
            int n = n0 + wn + nt * 16 + l16;
            float bias = bih[n] + bhh[n];
            #pragma unroll
            for (int r = 0; r < 8; r++) {
                int m = m0 + wm + mt * 16 + r + 8 * lhalf;
                int b = m >> 9;
                int s = m & 511;
                __builtin_nontemporal_store(
                    acc[mt][nt][r] + bias,
                    &xg[(((size_t)dir * SSZ + s) * BSZ + b) * G4H + n]);
            }
        }
}

// ---------------------------------------------------------------------------
// Kernel R: persistent recurrence. grid=2 (dir), 768 threads = 24 waves.
// Wave w owns h-columns [16w, 16w+16) across all 4 gates and both 16-row
// batch tiles. h lives in LDS, c in VGPRs. xg[t+1] is prefetched during the
// step-t WMMA chain (global_prefetch_b8) to hide the 192KB/step stream.
// out addr = dir*dirStride + (b*512+t)*rowStride + n
// ---------------------------------------------------------------------------
__global__ __launch_bounds__(768)
void lstm_recur_kernel(const float* __restrict__ xg,
                       const float* __restrict__ Whhf, const float* __restrict__ Whhb,
                       float* __restrict__ out, size_t dirStride, int rowStride)
{
    const int dir = blockIdx.x;
    const float* __restrict__ Whh = dir ? Whhb : Whhf;
    const float* __restrict__ xgd = xg + (size_t)dir * SSZ * BSZ * G4H;

    __shared__ float hs[32][HDIM + 1];   // stride 385 -> conflict-free frag reads

    const int tid   = threadIdx.x;
    const int lane  = tid & 31;
    const int wid   = tid >> 5;      // 0..23
    const int hc0   = wid * 16;
    const int lhalf = lane >> 4;
    const int l16   = lane & 15;

    for (int i = tid; i < 32 * (HDIM + 1); i += 768)
        (&hs[0][0])[i] = 0.0f;

    v8f cst[2] = {};   // cell state, matches C/D tile layout
    float* outd = out + dir * dirStride;

    __syncthreads();

    for (int t = 0; t < SSZ; t++) {
        // init gate accumulators from precomputed xg (contiguous 192KB block;
        // prefetched during the previous step's GEMM -> cache hits here)
        v8f acc[4][2];
        const float* __restrict__ xgt = xgd + (size_t)t * BSZ * G4H;
        #pragma unroll
        for (int g = 0; g < 4; g++)
            #pragma unroll
            for (int mt = 0; mt < 2; mt++)
                #pragma unroll
                for (int r = 0; r < 8; r++) {
                    int b = mt * 16 + r + 8 * lhalf;
                    acc[g][mt][r] = xgt[(size_t)b * G4H + g * HDIM + hc0 + l16];
                }

        // prefetch next step's xg block: 192KB = 768 threads x 2 cachelines.
        // Issued before the WMMA chain so the DMA overlaps the math.
        if (t + 1 < SSZ) {
            const char* nxt = (const char*)(xgd + (size_t)(t + 1) * BSZ * G4H);
            __builtin_prefetch(nxt + (size_t)tid * 256, 0, 0);
            __builtin_prefetch(nxt + (size_t)tid * 256 + 128, 0, 0);
        }

        // gates += h @ Whh^T  (K = 384, f32 WMMA, K-step 4)
        for (int kc = 0; kc < HDIM; kc += 4) {
            v2f af[2];
            #pragma unroll
            for (int mt = 0; mt < 2; mt++) {
                int b = mt * 16 + l16;
                af[mt].x = hs[b][kc + 2 * lhalf];
                af[mt].y = hs[b][kc + 2 * lhalf + 1];
            }
            #pragma unroll
            for (int g = 0; g < 4; g++) {
                int n = g * HDIM + hc0 + l16;
                const float* wp = Whh + (size_t)n * HDIM + kc + 2 * lhalf;
                v2f bf;
                bf.x = wp[0];
                bf.y = wp[1];
                acc[g][0] = WMMA_F32(af[0], bf, acc[g][0]);
                acc[g][1] = WMMA_F32(af[1], bf, acc[g][1]);
            }
        }
        __syncthreads();   // all reads of hs for this step done

        // elementwise LSTM cell; write new h to LDS + global (streaming -> NT)
        #pragma unroll
        for (int mt = 0; mt < 2; mt++) {
            #pragma unroll
            for (int r = 0; r < 8; r++) {
                float iv = sigmoid_f(acc[0][mt][r]);
                float fv = sigmoid_f(acc[1][mt][r]);
                float gv = tanh_f(acc[2][mt][r]);
                float ov = sigmoid_f(acc[3][mt][r]);
                float c  = fv * cst[mt][r] + iv * gv;
                cst[mt][r] = c;
                float h  = ov * tanh_f(c);
                int b = mt * 16 + r + 8 * lhalf;
                int n = hc0 + l16;
                hs[b][n] = h;
                __builtin_nontemporal_store(
                    h, &outd[(size_t)(b * SSZ + t) * rowStride + n]);
            }
        }
        __syncthreads();   // h published before next step's frag loads
    }
}

// ---------------------------------------------------------------------------
// Launch
// ---------------------------------------------------------------------------
extern "C" void kernel_launch(void* const* d_in, const int* in_sizes, int n_in,
                              void* d_out, int out_size, void* d_ws, size_t ws_size,
                              hipStream_t stream) {
    const float* emb    = (const float*)d_in[0];
    // d_in[1] = seg (unused by reference math)
    const float* wih_f0 = (const float*)d_in[2];
    const float* whh_f0 = (const float*)d_in[3];
    const float* bih_f0 = (const float*)d_in[4];
    const float* bhh_f0 = (const float*)d_in[5];
    const float* wih_f1 = (const float*)d_in[6];
    const float* whh_f1 = (const float*)d_in[7];
    const float* bih_f1 = (const float*)d_in[8];
    const float* bhh_f1 = (const float*)d_in[9];
    const float* wih_b0 = (const float*)d_in[10];
    const float* whh_b0 = (const float*)d_in[11];
    const float* bih_b0 = (const float*)d_in[12];
    const float* bhh_b0 = (const float*)d_in[13];
    const float* wih_b1 = (const float*)d_in[14];
    const float* whh_b1 = (const float*)d_in[15];
    const float* bih_b1 = (const float*)d_in[16];
    const float* bhh_b1 = (const float*)d_in[17];

    float* xg = (float*)d_ws;                                   // [2][512][32][1536]
    float* h0 = xg + (size_t)2 * SSZ * BSZ * G4H;               // [2][16384][384]
    float* out = (float*)d_out;                                 // [32][512][768]

    dim3 gA(G4H / 64, ROWS / 64, 2);   // (24, 256, 2)
    dim3 bA(128);
    dim3 gR(2);
    dim3 bR(768);

    // layer 0: xg from emb (backward dir reads time-reversed)
    xg_gemm_kernel<<<gA, bA, 0, stream>>>(emb, emb, wih_f0, wih_b0,
                                          bih_f0, bhh_f0, bih_b0, bhh_b0,
                                          xg, EDIM, 1);
    // layer 0 recurrence -> h0[dir][b*512+t][384]
    lstm_recur_kernel<<<gR, bR, 0, stream>>>(xg, whh_f0, whh_b0,
                                             h0, (size_t)ROWS * HDIM, HDIM);
    // layer 1: xg from h0 (no reversal)
    xg_gemm_kernel<<<gA, bA, 0, stream>>>(h0, h0 + (size_t)ROWS * HDIM,
                                          wih_f1, wih_b1,
                                          bih_f1, bhh_f1, bih_b1, bhh_b1,
                                          xg, HDIM, 0);
    // layer 1 recurrence -> d_out[b][t][dir*384 + n]
    lstm_recur_kernel<<<gR, bR, 0, stream>>>(xg, whh_f1, whh_b1,
                                             out, (size_t)HDIM, 2 * HDIM);
}